// OptimizedQuantumEmbedding_75728863363350
// MI455X (gfx1250) — compile-verified
//
#include <hip/hip_runtime.h>
#include <hip/hip_bf16.h>
#include <math.h>

#define NQ    6
#define NL    3
#define VOCAB 32000
#define EMB   512
#define BSEQ  (8 * 8192)   // B*S = 65536 tokens

typedef __attribute__((ext_vector_type(2))) float v2f;
typedef __attribute__((ext_vector_type(4))) float v4f;
typedef __attribute__((ext_vector_type(8))) float v8f;

// Branchless tanh: prefer CDNA5 hardware v_tanh_f32.
__device__ __forceinline__ float fast_tanh(float x)
{
#if __has_builtin(__builtin_amdgcn_tanhf)
    return __builtin_amdgcn_tanhf(x);
#else
    // tanh(x) = sign(x) * (1 - e) / (1 + e), e = exp(-2|x|); branchless v_exp.
    float ax = __builtin_fabsf(x);
    float e  = __expf(-2.0f * ax);
    float r  = (1.0f - e) / (1.0f + e);
    return __builtin_copysignf(r, x);
#endif
}

// ---------------------------------------------------------------------------
// Kernel 1: quantum feature precompute. One thread simulates one token's
// 6-qubit circuit with the 64-amplitude statevector held entirely in VGPRs
// (all state indices compile-time via full unroll). ~0.3 GFLOP total: trivial.
// Writes feats[v*8 + q] (stride 8 so fragment loads are 8B/16B aligned).
// ---------------------------------------------------------------------------
__global__ __launch_bounds__(256)
void qfeat_kernel(const float* __restrict__ params, float* __restrict__ feats)
{
    int v = blockIdx.x * blockDim.x + threadIdx.x;
    if (v >= VOCAB) return;
    const float* P = params + (size_t)v * (NL * NQ * 2);

    float sr[64], si[64];
#pragma unroll
    for (int i = 0; i < 64; ++i) { sr[i] = 0.f; si[i] = 0.f; }
    sr[0] = 1.f;

    for (int layer = 0; layer < NL; ++layer) {
#pragma unroll
        for (int q = 0; q < NQ; ++q) {
            float hy = 0.5f * P[layer * NQ * 2 + q * 2 + 0];
            float hz = 0.5f * P[layer * NQ * 2 + q * 2 + 1];
            float cy = cosf(hy), sy = sinf(hy);
            float cz = cosf(hz), sz = sinf(hz);
            const int st = 32 >> q;   // qubit q <-> bit (5-q)
#pragma unroll
            for (int i = 0; i < 64; ++i) {
                if ((i & st) == 0) {
                    const int j = i + st;
                    float t0r = cy * sr[i] - sy * sr[j];
                    float t0i = cy * si[i] - sy * si[j];
                    float t1r = sy * sr[i] + cy * sr[j];
                    float t1i = sy * si[i] + cy * si[j];
                    sr[i] = cz * t0r + sz * t0i;   // e^{-i hz}
                    si[i] = cz * t0i - sz * t0r;
                    sr[j] = cz * t1r - sz * t1i;   // e^{+i hz}
                    si[j] = cz * t1i + sz * t1r;
                }
            }
        }
        if (layer < NL - 1) {
#pragma unroll
            for (int q = 0; q < NQ - 1; ++q) {
                const int cs = 32 >> q, ts = 16 >> q;
#pragma unroll
                for (int i = 0; i < 64; ++i) {
                    if ((i & cs) && !(i & ts)) {
                        const int j = i | ts;
                        float tr = sr[i]; sr[i] = sr[j]; sr[j] = tr;
                        float ti = si[i]; si[i] = si[j]; si[j] = ti;
                    }
                }
            }
        }
    }

    float z[NQ];
#pragma unroll
    for (int q = 0; q < NQ; ++q) z[q] = 0.f;
#pragma unroll
    for (int i = 0; i < 64; ++i) {
        float p = sr[i] * sr[i] + si[i] * si[i];
#pragma unroll
        for (int q = 0; q < NQ; ++q)
            z[q] += ((i >> (5 - q)) & 1) ? -p : p;
    }

    float* F = feats + (size_t)v * 8;
#pragma unroll
    for (int q = 0; q < NQ; ++q) F[q] = z[q];
    F[6] = 0.f; F[7] = 0.f;
}

// ---------------------------------------------------------------------------
// Kernel 2: gather + GEMM(K=6 padded to 8) + bias + tanh via
// V_WMMA_F32_16X16X4_F32, with M = embedding dim, N = token dim:
//   D[m=emb][n=tok] = sum_k W[emb][k] * feat[tok][k]
//
// f32 16x16x4 fragment layout (ISA 7.12.2):
//   A (16x4, 2 VGPRs): lane<16 -> M=lane,  {V0,V1}={K0,K1};
//                      lane>=16-> M=lane-16,{V0,V1}={K2,K3}
//   B (4x16) mirrors A with N in place of M.
//   C/D (16x16, 8 VGPRs): VGPR r: lane<16 -> (M=r, N=lane),
//                                 lane>=16-> (M=r+8, N=lane-16)
// => each lane holds 8 CONSECUTIVE embedding values of one token:
//    out[tok][ebase + 8*half + 0..7]  -> two aligned b128 stores.
//
// One wave owns 16 tokens (B fragment built once, gather included) and walks
// the 32 embedding tiles streaming W (12 KB, WGP$-resident) through A.
// ---------------------------------------------------------------------------
__global__ __launch_bounds__(256)
void embed_wmma_kernel(const int*   __restrict__ tok,
                       const float* __restrict__ feats,
                       const float* __restrict__ W,     // [EMB][NQ] row-major
                       const float* __restrict__ bias,  // [EMB]
                       float*       __restrict__ out)   // [BSEQ][EMB]
{
    const int lane  = threadIdx.x & 31;
    const int wave  = threadIdx.x >> 5;
    const int ttile = blockIdx.x * 8 + wave;   // 16 tokens per wave
    const int half  = lane >> 4;               // 0: lanes 0-15, 1: lanes 16-31
    const int mn    = lane & 15;
    const float zf  = (half == 0) ? 1.0f : 0.0f;  // mask for K={4,5} vs {6,7}=0

    __builtin_prefetch(W, 0, 3);
    __builtin_prefetch(bias, 0, 3);

    // B fragments: token features (gathered once per wave), K padded 6 -> 8.
    const int row = ttile * 16 + mn;
    const int tid = tok[row];
    const float* F = feats + (size_t)tid * 8;
    v2f b0 = *(const v2f*)(F + half * 2);      // half0: K{0,1}; half1: K{2,3}
    v2f bt = *(const v2f*)(F + 4);             // K{4,5}
    v2f b1; b1.x = bt.x * zf; b1.y = bt.y * zf;   // half1 -> K{6,7} = 0

    for (int et = 0; et < EMB / 16; ++et) {
        const int m = et * 16 + mn;            // embedding row for A fragment
        const float* Wm = W + (size_t)m * NQ;
        v2f a0 = *(const v2f*)(Wm + half * 2);
        v2f at = *(const v2f*)(Wm + 4);
        v2f a1; a1.x = at.x * zf; a1.y = at.y * zf;

        v8f acc = {};
        acc = __builtin_amdgcn_wmma_f32_16x16x4_f32(
                  false, a0, false, b0, (short)0, acc, false, false);
        acc = __builtin_amdgcn_wmma_f32_16x16x4_f32(
                  false, a1, false, b1, (short)0, acc, false, false);

        const int ebase = et * 16 + half * 8;  // this lane's 8 consecutive embs
        const v4f bv0 = *(const v4f*)(bias + ebase);
        const v4f bv1 = *(const v4f*)(bias + ebase + 4);

        v4f lo, hi;
        lo.x = fast_tanh(acc[0] + bv0.x);
        lo.y = fast_tanh(acc[1] + bv0.y);
        lo.z = fast_tanh(acc[2] + bv0.z);
        lo.w = fast_tanh(acc[3] + bv0.w);
        hi.x = fast_tanh(acc[4] + bv1.x);
        hi.y = fast_tanh(acc[5] + bv1.y);
        hi.z = fast_tanh(acc[6] + bv1.z);
        hi.w = fast_tanh(acc[7] + bv1.w);

        float* o = out + (size_t)row * EMB + ebase;
        *(v4f*)o       = lo;
        *(v4f*)(o + 4) = hi;
    }
}

// ---------------------------------------------------------------------------
extern "C" void kernel_launch(void* const* d_in, const int* in_sizes, int n_in,
                              void* d_out, int out_size, void* d_ws, size_t ws_size,
                              hipStream_t stream)
{
    const int*   token_ids = (const int*)  d_in[0];  // [8, 8192] int32
    const float* qparams   = (const float*)d_in[1];  // [VOCAB, 36] f32
    const float* W         = (const float*)d_in[2];  // [EMB, NQ] f32
    const float* b         = (const float*)d_in[3];  // [EMB] f32
    float*       out       = (float*)d_out;          // [8, 8192, EMB] f32

    float* feats = (float*)d_ws;                     // VOCAB*8 f32 = 1 MB

    qfeat_kernel<<<(VOCAB + 255) / 256, 256, 0, stream>>>(qparams, feats);

    // 65536 tokens / 16 per wave = 4096 waves; 8 waves (256 thr) per block
    embed_wmma_kernel<<<(BSEQ / 16) / 8, 256, 0, stream>>>(
        token_ids, feats, W, b, out);
}